// CausalDisentanglementModule_58506044506295
// MI455X (gfx1250) — compile-verified
//
#include <hip/hip_runtime.h>
#include <hip/hip_bf16.h>
#include <stdint.h>

#define N_USERS 150000
#define N_ITEMS 50000
#define N_TOTAL (N_USERS + N_ITEMS)
#define EMBED_DIM 64
#define N_FACTORS 4
#define FACTOR_DIM 16
#define D2 32
#define HIDDEN 64
#define NNZ 3200000
#define N_LAYERS 2

typedef __attribute__((ext_vector_type(2))) float v2f;
typedef __attribute__((ext_vector_type(8))) float v8f;

// ---------------------------------------------------------------- RNG helpers
__device__ __forceinline__ uint32_t pcg_hash(uint32_t x) {
  x = x * 747796405u + 2891336453u;
  uint32_t w = ((x >> ((x >> 28) + 4u)) ^ x) * 277803737u;
  return (w >> 22) ^ w;
}

__device__ __forceinline__ float gaussian(uint32_t s) {
  uint32_t a = pcg_hash(s);
  uint32_t b = pcg_hash(s ^ 0x9E3779B9u);
  float u1 = ((a >> 8) + 1u) * (1.0f / 16777216.0f);  // (0,1]
  float u2 = (b >> 8) * (1.0f / 16777216.0f);         // [0,1)
  return sqrtf(-2.0f * __logf(u1)) * __cosf(6.28318530718f * u2);
}

// ------------------------------------------------- concat tables, seed accum
__global__ __launch_bounds__(256) void init_kernel(
    const float* __restrict__ user_table, const float* __restrict__ item_table,
    float* __restrict__ cur, float* __restrict__ sum) {
  size_t tid = (size_t)blockIdx.x * 256u + threadIdx.x;
  const size_t uelems = (size_t)N_USERS * EMBED_DIM;
  float v = (tid < uelems) ? user_table[tid] : item_table[tid - uelems];
  cur[tid] = v;   // layer-0 input
  sum[tid] = v;   // running sum for the final mean (in d_out)
}

// ----------------------------------------------- SpMM: gather + scatter-add
// thread t: edge = t/16, handles 4 dims via float4 gather + 4 f32 atomics.
__global__ __launch_bounds__(256) void spmm_kernel(
    const int* __restrict__ rows, const int* __restrict__ cols,
    const float* __restrict__ vals, const float* __restrict__ cur,
    float* __restrict__ msgs) {
  unsigned tid = blockIdx.x * 256u + threadIdx.x;
  unsigned e = tid >> 4;
  unsigned part = tid & 15u;
  int r = rows[e];
  int c = cols[e];
  float v = vals[e];
  float4 x = *((const float4*)(cur + (size_t)c * EMBED_DIM) + part);
  float* dst = msgs + (size_t)r * EMBED_DIM + part * 4u;
  atomicAdd(dst + 0, v * x.x);
  atomicAdd(dst + 1, v * x.y);
  atomicAdd(dst + 2, v * x.z);
  atomicAdd(dst + 3, v * x.w);
}

// -------------------------------------------- per-factor VAE encoder (WMMA)
// Block = 128 threads = 4 waves; wave f handles factor f for a 16-node tile.
// Both encoders (c and s) run per wave; z's combined by sigmoid gate.
__global__ __launch_bounds__(128) void mlp_kernel(
    const float* __restrict__ cur, const float* __restrict__ msgs,
    const float* __restrict__ Wc1, const float* __restrict__ bc1,
    const float* __restrict__ Wc2, const float* __restrict__ bc2,
    const float* __restrict__ Ws1, const float* __restrict__ bs1,
    const float* __restrict__ Ws2, const float* __restrict__ bs2,
    const float* __restrict__ gates,
    float* __restrict__ cur_out, float* __restrict__ sum_out, int layer) {
  // hidden activations per factor, padded stride 65 -> conflict-free banks
  __shared__ float h_lds[N_FACTORS][16][HIDDEN + 1];

  const int lane = threadIdx.x & 31;
  const int f = threadIdx.x >> 5;      // factor handled by this wave
  const int mlo = lane & 15;           // M (A/C rows) or N (B/C cols) low index
  const int khalf = lane >> 4;         // selects K pair {0,1} vs {2,3}
  const int node0 = blockIdx.x * 16;

  // A fragments for combined[16 x 32]: K 0..15 = cur slice, 16..31 = msgs slice
  // 16x4 f32 A layout: lanes 0-15 hold K=k0,k0+1 ; lanes 16-31 hold K=k0+2,k0+3
  v2f afrag[8];
  {
    const size_t nb = (size_t)(node0 + mlo) * EMBED_DIM + f * FACTOR_DIM;
    const float* cb = cur + nb;
    const float* mb = msgs + nb;
#pragma unroll
    for (int k = 0; k < 4; ++k) {
      int k0 = 4 * k + 2 * khalf;
      v2f ac = {cb[k0], cb[k0 + 1]};
      v2f am = {mb[k0], mb[k0 + 1]};
      afrag[k] = ac;
      afrag[k + 4] = am;
    }
  }

  float z[2][8];

#pragma unroll
  for (int enc = 0; enc < 2; ++enc) {
    const float* W1 = (enc == 0 ? Wc1 : Ws1) + f * D2 * HIDDEN;   // [32][64]
    const float* B1 = (enc == 0 ? bc1 : bs1) + f * HIDDEN;        // [64]
    const float* W2 = (enc == 0 ? Wc2 : Ws2) + f * HIDDEN * D2;   // [64][32]
    const float* B2 = (enc == 0 ? bc2 : bs2) + f * D2;            // [32]

    // ---- GEMM1: h[16x64] = relu(A[16x32] @ W1[32x64] + b1)
#pragma unroll
    for (int j = 0; j < 4; ++j) {  // 16-wide output column tiles
      v8f acc = {0.f, 0.f, 0.f, 0.f, 0.f, 0.f, 0.f, 0.f};
#pragma unroll
      for (int k = 0; k < 8; ++k) {  // K in steps of 4
        int k0 = 4 * k + 2 * khalf;
        const float* wp = W1 + k0 * HIDDEN + 16 * j + mlo;
        v2f b = {wp[0], wp[HIDDEN]};
        acc = __builtin_amdgcn_wmma_f32_16x16x4_f32(
            false, afrag[k], false, b, (short)0, acc, false, false);
      }
      float bias = B1[16 * j + mlo];
#pragma unroll
      for (int r = 0; r < 8; ++r) {  // D row M = r + 8*khalf, col N = mlo
        float hv = acc[r] + bias;
        h_lds[f][r + 8 * khalf][16 * j + mlo] = hv > 0.f ? hv : 0.f;
      }
    }
    __syncthreads();

    // ---- GEMM2: p[16x32] = h[16x64] @ W2[64x32] + b2 ; cols 0-15 mu, 16-31 lv
    v8f pm = {0.f, 0.f, 0.f, 0.f, 0.f, 0.f, 0.f, 0.f};
    v8f pl = {0.f, 0.f, 0.f, 0.f, 0.f, 0.f, 0.f, 0.f};
#pragma unroll
    for (int k = 0; k < 16; ++k) {
      int k0 = 4 * k + 2 * khalf;
      v2f a = {h_lds[f][mlo][k0], h_lds[f][mlo][k0 + 1]};
      const float* wp = W2 + k0 * D2 + mlo;
      v2f bm = {wp[0], wp[D2]};
      v2f bl = {wp[16], wp[16 + D2]};
      pm = __builtin_amdgcn_wmma_f32_16x16x4_f32(
          false, a, false, bm, (short)0, pm, false, false);
      pl = __builtin_amdgcn_wmma_f32_16x16x4_f32(
          false, a, false, bl, (short)0, pl, false, false);
    }
    float bmu = B2[mlo];
    float blv = B2[16 + mlo];
#pragma unroll
    for (int r = 0; r < 8; ++r) {
      int node = node0 + r + 8 * khalf;
      float mu = pm[r] + bmu;
      float lv = pl[r] + blv;
      uint32_t seed =
          ((uint32_t)node * 64u + (uint32_t)f * 16u + (uint32_t)mlo) * 2654435761u ^
          ((uint32_t)(2 * layer + enc) * 0x85EBCA6Bu + 0x01234567u);
      z[enc][r] = mu + gaussian(seed) * __expf(0.5f * lv);
    }
    __syncthreads();
  }

  // gate and emit: fused = g*z_c + (1-g)*z_s ; accumulate into mean sum
  float g = 1.0f /
            (1.0f + __expf(-gates[layer * (N_FACTORS * FACTOR_DIM) +
                                  f * FACTOR_DIM + mlo]));
#pragma unroll
  for (int r = 0; r < 8; ++r) {
    int node = node0 + r + 8 * khalf;
    float fused = g * z[0][r] + (1.0f - g) * z[1][r];
    size_t idx = (size_t)node * EMBED_DIM + f * FACTOR_DIM + mlo;
    cur_out[idx] = fused;
    sum_out[idx] += fused;  // each (node,f,d) touched by exactly one lane
  }
}

// ---------------------------------------------------------------- mean scale
__global__ __launch_bounds__(256) void finalize_kernel(float* __restrict__ out) {
  size_t tid = (size_t)blockIdx.x * 256u + threadIdx.x;
  out[tid] *= (1.0f / 3.0f);
}

// ---------------------------------------------------------------------------
extern "C" void kernel_launch(void* const* d_in, const int* in_sizes, int n_in,
                              void* d_out, int out_size, void* d_ws, size_t ws_size,
                              hipStream_t stream) {
  // setup_inputs() order:
  // 0 user_table, 1 item_table, 2 rows, 3 cols, 4 vals,
  // 5 Wc1, 6 bc1, 7 Wc2, 8 bc2, 9 Ws1, 10 bs1, 11 Ws2, 12 bs2,
  // 13 gates, 14 n_users
  const float* user_table = (const float*)d_in[0];
  const float* item_table = (const float*)d_in[1];
  const int* rows = (const int*)d_in[2];
  const int* cols = (const int*)d_in[3];
  const float* vals = (const float*)d_in[4];
  const float* Wc1 = (const float*)d_in[5];
  const float* bc1 = (const float*)d_in[6];
  const float* Wc2 = (const float*)d_in[7];
  const float* bc2 = (const float*)d_in[8];
  const float* Ws1 = (const float*)d_in[9];
  const float* bs1 = (const float*)d_in[10];
  const float* Ws2 = (const float*)d_in[11];
  const float* bs2 = (const float*)d_in[12];
  const float* gates = (const float*)d_in[13];
  (void)in_sizes; (void)n_in; (void)out_size; (void)ws_size;

  const size_t nelem = (size_t)N_TOTAL * EMBED_DIM;  // 12.8M floats
  float* curA = (float*)d_ws;
  float* curB = curA + nelem;
  float* msgs = curB + nelem;
  float* sum = (float*)d_out;  // running sum of {emb0, emb1, emb2}

  init_kernel<<<(unsigned)(nelem / 256), 256, 0, stream>>>(user_table, item_table,
                                                           curA, sum);

  const float* cur_in = curA;
  float* cur_out = curB;
  for (int layer = 0; layer < N_LAYERS; ++layer) {
    (void)hipMemsetAsync(msgs, 0, nelem * sizeof(float), stream);
    spmm_kernel<<<(unsigned)(((size_t)NNZ * 16) / 256), 256, 0, stream>>>(
        rows, cols, vals, cur_in, msgs);
    mlp_kernel<<<N_TOTAL / 16, 128, 0, stream>>>(
        cur_in, msgs, Wc1, bc1, Wc2, bc2, Ws1, bs1, Ws2, bs2, gates,
        cur_out, sum, layer);
    float* t = (float*)cur_in;
    cur_in = cur_out;
    cur_out = t;
  }

  finalize_kernel<<<(unsigned)(nelem / 256), 256, 0, stream>>>(sum);
}